// CelestialPetriNetCombiner_14224931684714
// MI455X (gfx1250) — compile-verified
//
#include <hip/hip_runtime.h>

// ---------------------------------------------------------------------------
// CelestialPetriNetCombiner for MI455X (gfx1250), fp32 throughout.
// Matrix work on V_WMMA_F32_16X16X4_F32 (exact fp32 WMMA path, wave32).
// ---------------------------------------------------------------------------

typedef float v2f __attribute__((ext_vector_type(2)));
typedef float v8f __attribute__((ext_vector_type(8)));

constexpr int cB = 4, cS = 512, cN = 13, cD = 128, cE = 8, cM = 64, cH = 8, cDH = 16;
constexpr int cBS  = cB * cS;              // 2048
constexpr int cBN  = cB * cN;              // 52
constexpr int cNE  = cB * cS * cN * cN;    // 346112 edges
constexpr int cNSD = cB * cS * cN * cD;    // 3407872 node-state floats
constexpr int cND  = cN * cD;              // 1664

__device__ __forceinline__ float gelu_erf(float x) {
    return 0.5f * x * (1.0f + erff(x * 0.70710678118654752f));
}
__device__ __forceinline__ float sigm(float x) { return 1.0f / (1.0f + expf(-x)); }

__device__ __forceinline__ v8f wmma4(v2f a, v2f b, v8f c) {
    // D = A(16x4 f32) * B(4x16 f32) + C(16x16 f32)
    return __builtin_amdgcn_wmma_f32_16x16x4_f32(false, a, false, b, (short)0, c,
                                                 false, false);
}

// A: row-major [16 x lda] (M rows), B: row-major [K x ldb] (cols = N tile of 16).
template <typename PA, typename PB>
__device__ __forceinline__ v8f wmma_gemm(PA A, int lda, PB B, int ldb, int K,
                                         v8f c, int l16, int half) {
#pragma unroll 4
    for (int k0 = 0; k0 < K; k0 += 4) {
        const int ka = k0 + 2 * half;
        v2f a, b;
        a.x = A[l16 * lda + ka];
        a.y = A[l16 * lda + ka + 1];
        b.x = B[ka * ldb + l16];
        b.y = B[(ka + 1) * ldb + l16];
        c = wmma4(a, b, c);
    }
    return c;
}

// ---------------------------------------------------------------------------
// ctx = enc_out @ w1_ctx   [B*S, 16]
__global__ void k_ctx(const float* __restrict__ enc, const float* __restrict__ w,
                      float* __restrict__ ctx) {
    int gid = blockIdx.x * blockDim.x + threadIdx.x;
    if (gid >= cBS * 16) return;
    int bs = gid >> 4, c = gid & 15;
    float acc = 0.f;
    for (int d = 0; d < cD; ++d) acc += enc[bs * cD + d] * w[d * 16 + c];
    ctx[gid] = acc;
}

// ---------------------------------------------------------------------------
// per-edge: encode (LN+GELU, LN) then fusion -> enriched [.,8]
__global__ void k_edge(const float* __restrict__ ae, const float* __restrict__ le,
                       const float* __restrict__ te, const float* __restrict__ ctx,
                       const float* __restrict__ ew1, const float* __restrict__ eb1,
                       const float* __restrict__ eg1, const float* __restrict__ ebt1,
                       const float* __restrict__ ew2, const float* __restrict__ eb2,
                       const float* __restrict__ eg2, const float* __restrict__ ebt2,
                       const float* __restrict__ fwe, const float* __restrict__ fb1,
                       const float* __restrict__ fw2, const float* __restrict__ fb2,
                       float* __restrict__ enr) {
    int e = blockIdx.x * blockDim.x + threadIdx.x;
    if (e >= cNE) return;
    int bs = e / (cN * cN);
    float a = ae[e], l = le[e], t = te[e];
    float h[16];
    float mu = 0.f;
#pragma unroll
    for (int j = 0; j < 16; ++j) {
        h[j] = a * ew1[j] + l * ew1[16 + j] + t * ew1[32 + j] + eb1[j];
        mu += h[j];
    }
    mu *= (1.f / 16.f);
    float var = 0.f;
#pragma unroll
    for (int j = 0; j < 16; ++j) { float d = h[j] - mu; var += d * d; }
    float rstd = rsqrtf(var * (1.f / 16.f) + 1e-5f);
#pragma unroll
    for (int j = 0; j < 16; ++j)
        h[j] = gelu_erf((h[j] - mu) * rstd * eg1[j] + ebt1[j]);

    float ef[8];
    mu = 0.f;
#pragma unroll
    for (int k = 0; k < 8; ++k) {
        float acc = eb2[k];
#pragma unroll
        for (int j = 0; j < 16; ++j) acc += h[j] * ew2[j * 8 + k];
        ef[k] = acc; mu += acc;
    }
    mu *= (1.f / 8.f);
    var = 0.f;
#pragma unroll
    for (int k = 0; k < 8; ++k) { float d = ef[k] - mu; var += d * d; }
    rstd = rsqrtf(var * (1.f / 8.f) + 1e-5f);
#pragma unroll
    for (int k = 0; k < 8; ++k) ef[k] = (ef[k] - mu) * rstd * eg2[k] + ebt2[k];

    float f[16];
#pragma unroll
    for (int j = 0; j < 16; ++j) {
        float acc = ctx[bs * 16 + j] + fb1[j];
#pragma unroll
        for (int k = 0; k < 8; ++k) acc += ef[k] * fwe[k * 16 + j];
        f[j] = gelu_erf(acc);
    }
#pragma unroll
    for (int k = 0; k < 8; ++k) {
        float acc = fb2[k];
#pragma unroll
        for (int j = 0; j < 16; ++j) acc += f[j] * fw2[j * 8 + k];
        enr[(size_t)e * 8 + k] = acc;
    }
}

// ---------------------------------------------------------------------------
__global__ void k_init_ns(const float* __restrict__ enc, float* __restrict__ ns) {
    int gid = blockIdx.x * blockDim.x + threadIdx.x;
    if (gid >= cNSD) return;
    int bs = gid / cND;
    ns[gid] = enc[bs * cD + (gid & 127)];
}

// ---------------------------------------------------------------------------
// One message-passing round; one block per (b,s), 256 threads (8 waves).
__global__ void k_mp(const float* __restrict__ enr, float* __restrict__ ns,
                     const float* __restrict__ w_msg, const float* __restrict__ b_msg,
                     const float* __restrict__ wg_i, const float* __restrict__ wg_j,
                     const float* __restrict__ wg_e, const float* __restrict__ bg,
                     const float* __restrict__ wu_n, const float* __restrict__ wu_m,
                     const float* __restrict__ bu, const float* __restrict__ g,
                     const float* __restrict__ bb) {
    __shared__ float enr_s[169 * 8];
    __shared__ float gi_s[13], gj_s[13], gate_s[169];
    __shared__ float agg_s[16 * 64];
    __shared__ float nss[16 * 128];
    __shared__ float ys[16 * 128];
    const int bs = blockIdx.x;
    const int t = threadIdx.x;
    const int wave = t >> 5, lane = t & 31, half = lane >> 4, l16 = lane & 15;

    for (int idx = t; idx < 169 * 8; idx += 256)
        enr_s[idx] = enr[(size_t)bs * (169 * 8) + idx];
    for (int idx = t; idx < 16 * 128; idx += 256) {
        int i = idx >> 7;
        nss[idx] = (i < 13) ? ns[(size_t)bs * cND + idx] : 0.f;
    }
    for (int idx = t; idx < 16 * 64; idx += 256) agg_s[idx] = 0.f;
    __syncthreads();

    if (t < 26) {
        int i = t >> 1;
        const float* w = (t & 1) ? wg_j : wg_i;
        float acc = 0.f;
        for (int d = 0; d < 128; ++d) acc += nss[i * 128 + d] * w[d];
        if (t & 1) gj_s[i] = acc; else gi_s[i] = acc;
    }
    __syncthreads();
    if (t < 169) {
        int i = t / 13, j = t - i * 13;
        float s = gi_s[i] + gj_s[j] + bg[0];
#pragma unroll
        for (int e = 0; e < 8; ++e) s += enr_s[t * 8 + e] * wg_e[e];
        gate_s[t] = sigm(s);
    }
    __syncthreads();
    for (int idx = t; idx < 13 * 64; idx += 256) {
        int i = idx >> 6, mm = idx & 63;
        float acc = 0.f;
        for (int j = 0; j < 13; ++j) {
            const float* er = enr_s + (i * 13 + j) * 8;
            float msg = b_msg[mm];
#pragma unroll
            for (int e = 0; e < 8; ++e) msg += er[e] * w_msg[e * 64 + mm];
            acc += gate_s[i * 13 + j] * msg;
        }
        agg_s[i * 64 + mm] = acc;
    }
    __syncthreads();

    // node update GEMM: ns@wu_n (K=128) + agg@wu_m (K=64); one d-tile per wave
    {
        const int dt = wave;
        v8f c = wmma_gemm(nss, 128, wu_n + dt * 16, 128, 128, v8f{}, l16, half);
        c = wmma_gemm(agg_s, 64, wu_m + dt * 16, 128, 64, c, l16, half);
#pragma unroll
        for (int vv = 0; vv < 8; ++vv) {
            int r = vv + 8 * half, col = dt * 16 + l16;
            ys[r * 128 + col] = c[vv] + bu[col] + nss[r * 128 + col];
        }
    }
    __syncthreads();
    if (t < 13) {
        float mu = 0.f;
        for (int d = 0; d < 128; ++d) mu += ys[t * 128 + d];
        mu *= (1.f / 128.f);
        float var = 0.f;
        for (int d = 0; d < 128; ++d) { float dd = ys[t * 128 + d] - mu; var += dd * dd; }
        float rstd = rsqrtf(var * (1.f / 128.f) + 1e-5f);
        for (int d = 0; d < 128; ++d)
            ns[(size_t)bs * cND + t * 128 + d] =
                (ys[t * 128 + d] - mu) * rstd * g[d] + bb[d];
    }
}

// ---------------------------------------------------------------------------
// temporal QKV projections; block per (bn, s_tile) — x[s,:] = ns[b,s,n,:]
__global__ void k_tqkv(const float* __restrict__ ns,
                       const float* __restrict__ wq, const float* __restrict__ bq,
                       const float* __restrict__ wk, const float* __restrict__ bk,
                       const float* __restrict__ wv, const float* __restrict__ bv,
                       float* __restrict__ q, float* __restrict__ k,
                       float* __restrict__ v) {
    __shared__ float xs[16 * 128];
    const int bn = blockIdx.x >> 5, st = blockIdx.x & 31;
    const int b = bn / 13, n = bn - b * 13;
    const int t = threadIdx.x;
    const int wave = t >> 5, lane = t & 31, half = lane >> 4, l16 = lane & 15;
    for (int idx = t; idx < 16 * 128; idx += 256) {
        int r = idx >> 7, d = idx & 127, s = st * 16 + r;
        xs[idx] = ns[(((size_t)b * cS + s) * cN + n) * cD + d];
    }
    __syncthreads();
    const int dt = wave;
    v8f cq = wmma_gemm(xs, 128, wq + dt * 16, 128, 128, v8f{}, l16, half);
    v8f ck = wmma_gemm(xs, 128, wk + dt * 16, 128, 128, v8f{}, l16, half);
    v8f cv = wmma_gemm(xs, 128, wv + dt * 16, 128, 128, v8f{}, l16, half);
#pragma unroll
    for (int vv = 0; vv < 8; ++vv) {
        int r = vv + 8 * half, col = dt * 16 + l16, s = st * 16 + r;
        size_t o = ((size_t)bn * cS + s) * cD + col;
        q[o] = cq[vv] + bq[col];
        k[o] = ck[vv] + bk[col];
        v[o] = cv[vv] + bv[col];
    }
}

// ---------------------------------------------------------------------------
// temporal attention, flash-style online softmax; block per (bn, h)
__global__ void k_tattn(const float* __restrict__ q, const float* __restrict__ k,
                        const float* __restrict__ v, float* __restrict__ ao) {
    __shared__ float ks[256 * 16];
    __shared__ float vs[256 * 16];
    __shared__ float pb[8 * 256];
    const int bn = blockIdx.x >> 3, h = blockIdx.x & 7;
    const int t = threadIdx.x;
    const int wave = t >> 5, lane = t & 31, half = lane >> 4, l16 = lane & 15;
    const float* qh = q + (size_t)bn * cS * cD + h * 16;
    const float* kh = k + (size_t)bn * cS * cD + h * 16;
    const float* vh = v + (size_t)bn * cS * cD + h * 16;
    float* aoh = ao + (size_t)bn * cS * cD + h * 16;
    float* pbw = pb + wave * 256;

    for (int qc = 0; qc < 4; ++qc) {
        const int q0 = qc * 128 + wave * 16;
        v2f aq[4];
#pragma unroll
        for (int kk = 0; kk < 4; ++kk) {          // softmax scale folded into Q
            const int ka = kk * 4 + 2 * half;
            aq[kk].x = 0.25f * qh[(size_t)(q0 + l16) * cD + ka];
            aq[kk].y = 0.25f * qh[(size_t)(q0 + l16) * cD + ka + 1];
        }
        float m[8], l[8];
        v8f o = {};
#pragma unroll
        for (int vv = 0; vv < 8; ++vv) { m[vv] = -3.0e38f; l[vv] = 0.f; }
        for (int kc = 0; kc < 2; ++kc) {
            __syncthreads();
            for (int idx = t; idx < 256 * 16; idx += 256) {
                int s = idx >> 4, d = idx & 15, key = kc * 256 + s;
                ks[idx] = kh[(size_t)key * cD + d];
                vs[idx] = vh[(size_t)key * cD + d];
            }
            __syncthreads();
            for (int kt = 0; kt < 16; ++kt) {
                v8f c = {};
#pragma unroll
                for (int kk = 0; kk < 4; ++kk) {  // S = Q @ K^T, K-dim = 16
                    const int ka = kk * 4 + 2 * half;
                    v2f bfr;
                    bfr.x = ks[(kt * 16 + l16) * 16 + ka];
                    bfr.y = ks[(kt * 16 + l16) * 16 + ka + 1];
                    c = wmma4(aq[kk], bfr, c);
                }
#pragma unroll
                for (int vv = 0; vv < 8; ++vv) {  // online softmax, per-row stats
                    float rm = c[vv];
                    rm = fmaxf(rm, __shfl_xor(rm, 1));
                    rm = fmaxf(rm, __shfl_xor(rm, 2));
                    rm = fmaxf(rm, __shfl_xor(rm, 4));
                    rm = fmaxf(rm, __shfl_xor(rm, 8));
                    float mn = fmaxf(m[vv], rm);
                    float al = expf(m[vv] - mn);
                    float pe = expf(c[vv] - mn);
                    float rs = pe;
                    rs += __shfl_xor(rs, 1);
                    rs += __shfl_xor(rs, 2);
                    rs += __shfl_xor(rs, 4);
                    rs += __shfl_xor(rs, 8);
                    l[vv] = l[vv] * al + rs;
                    m[vv] = mn;
                    o[vv] *= al;
                    pbw[(vv + 8 * half) * 16 + l16] = pe;
                }
                __syncthreads();                  // relayout P (C-layout -> A-layout)
#pragma unroll
                for (int kk = 0; kk < 4; ++kk) {  // O += P @ V_tile
                    const int ka = kk * 4 + 2 * half;
                    v2f a, bfr;
                    a.x = pbw[l16 * 16 + ka];
                    a.y = pbw[l16 * 16 + ka + 1];
                    bfr.x = vs[(kt * 16 + ka) * 16 + l16];
                    bfr.y = vs[(kt * 16 + ka + 1) * 16 + l16];
                    o = wmma4(a, bfr, o);
                }
                __syncthreads();
            }
        }
#pragma unroll
        for (int vv = 0; vv < 8; ++vv)
            aoh[(size_t)(q0 + vv + 8 * half) * cD + l16] = o[vv] / l[vv];
    }
}

// ---------------------------------------------------------------------------
// temporal out-proj + residual + LN; writes back into ns layout [B,S,N,D]
__global__ void k_tout(const float* __restrict__ ao, float* __restrict__ ns,
                       const float* __restrict__ wo, const float* __restrict__ bo,
                       const float* __restrict__ g, const float* __restrict__ bb) {
    __shared__ float os[16 * 128];
    __shared__ float ys[16 * 128];
    const int bn = blockIdx.x >> 5, st = blockIdx.x & 31;
    const int b = bn / 13, n = bn - b * 13;
    const int t = threadIdx.x;
    const int wave = t >> 5, lane = t & 31, half = lane >> 4, l16 = lane & 15;
    for (int idx = t; idx < 16 * 128; idx += 256) {
        int r = idx >> 7, d = idx & 127, s = st * 16 + r;
        os[idx] = ao[((size_t)bn * cS + s) * cD + d];
    }
    __syncthreads();
    {
        const int dt = wave;
        v8f c = wmma_gemm(os, 128, wo + dt * 16, 128, 128, v8f{}, l16, half);
#pragma unroll
        for (int vv = 0; vv < 8; ++vv) {
            int r = vv + 8 * half, col = dt * 16 + l16, s = st * 16 + r;
            size_t xi = (((size_t)b * cS + s) * cN + n) * cD + col;
            ys[r * 128 + col] = c[vv] + bo[col] + ns[xi];
        }
    }
    __syncthreads();
    if (t < 16) {
        int s = st * 16 + t;
        float mu = 0.f;
        for (int d = 0; d < 128; ++d) mu += ys[t * 128 + d];
        mu *= (1.f / 128.f);
        float var = 0.f;
        for (int d = 0; d < 128; ++d) { float dd = ys[t * 128 + d] - mu; var += dd * dd; }
        float rstd = rsqrtf(var * (1.f / 128.f) + 1e-5f);
        for (int d = 0; d < 128; ++d)
            ns[(((size_t)b * cS + s) * cN + n) * cD + d] =
                (ys[t * 128 + d] - mu) * rstd * g[d] + bb[d];
    }
}

// ---------------------------------------------------------------------------
// spatial MHA fully fused per (b,s); final ns written to d_out region
__global__ void k_spatial(const float* __restrict__ ns,
                          const float* __restrict__ wq, const float* __restrict__ bq,
                          const float* __restrict__ wk, const float* __restrict__ bk,
                          const float* __restrict__ wv, const float* __restrict__ bv,
                          const float* __restrict__ wo, const float* __restrict__ bo,
                          const float* __restrict__ g, const float* __restrict__ bb,
                          float* __restrict__ out) {
    __shared__ float xs[16 * 128];
    __shared__ float qs[16 * 128];
    __shared__ float ks2[16 * 128];
    __shared__ float vs2[16 * 128];
    __shared__ float ps[8 * 256];
    __shared__ float ys[16 * 128];
    const int bs = blockIdx.x;
    const int t = threadIdx.x;
    const int wave = t >> 5, lane = t & 31, half = lane >> 4, l16 = lane & 15;
    for (int idx = t; idx < 16 * 128; idx += 256) {
        int i = idx >> 7;
        xs[idx] = (i < 13) ? ns[(size_t)bs * cND + idx] : 0.f;
    }
    __syncthreads();
    {   // QKV projections, one d-tile per wave
        const int dt = wave;
        v8f cq = wmma_gemm(xs, 128, wq + dt * 16, 128, 128, v8f{}, l16, half);
        v8f ck = wmma_gemm(xs, 128, wk + dt * 16, 128, 128, v8f{}, l16, half);
        v8f cv = wmma_gemm(xs, 128, wv + dt * 16, 128, 128, v8f{}, l16, half);
#pragma unroll
        for (int vv = 0; vv < 8; ++vv) {
            int r = vv + 8 * half, col = dt * 16 + l16;
            qs[r * 128 + col] = cq[vv] + bq[col];
            ks2[r * 128 + col] = ck[vv] + bk[col];
            vs2[r * 128 + col] = cv[vv] + bv[col];
        }
    }
    __syncthreads();
    {   // scores + masked softmax, one head per wave
        const int hh = wave;
        v8f c = {};
#pragma unroll
        for (int kk = 0; kk < 4; ++kk) {
            const int ka = kk * 4 + 2 * half;
            v2f a, bfr;
            a.x = 0.25f * qs[l16 * 128 + hh * 16 + ka];
            a.y = 0.25f * qs[l16 * 128 + hh * 16 + ka + 1];
            bfr.x = ks2[l16 * 128 + hh * 16 + ka];
            bfr.y = ks2[l16 * 128 + hh * 16 + ka + 1];
            c = wmma4(a, bfr, c);
        }
#pragma unroll
        for (int vv = 0; vv < 8; ++vv) {
            float sc = (l16 < 13) ? c[vv] : -3.0e38f;
            float rm = sc;
            rm = fmaxf(rm, __shfl_xor(rm, 1));
            rm = fmaxf(rm, __shfl_xor(rm, 2));
            rm = fmaxf(rm, __shfl_xor(rm, 4));
            rm = fmaxf(rm, __shfl_xor(rm, 8));
            float pe = expf(sc - rm);
            float rs = pe;
            rs += __shfl_xor(rs, 1);
            rs += __shfl_xor(rs, 2);
            rs += __shfl_xor(rs, 4);
            rs += __shfl_xor(rs, 8);
            ps[hh * 256 + (vv + 8 * half) * 16 + l16] = pe / rs;
        }
    }
    __syncthreads();
    {   // O_h = P_h @ V_h, written into qs (reused as concat buffer)
        const int hh = wave;
        v8f o = {};
#pragma unroll
        for (int kk = 0; kk < 4; ++kk) {
            const int ka = kk * 4 + 2 * half;
            v2f a, bfr;
            a.x = ps[hh * 256 + l16 * 16 + ka];
            a.y = ps[hh * 256 + l16 * 16 + ka + 1];
            bfr.x = vs2[ka * 128 + hh * 16 + l16];
            bfr.y = vs2[(ka + 1) * 128 + hh * 16 + l16];
            o = wmma4(a, bfr, o);
        }
        __syncthreads();
#pragma unroll
        for (int vv = 0; vv < 8; ++vv)
            qs[(vv + 8 * half) * 128 + hh * 16 + l16] = o[vv];
    }
    __syncthreads();
    {   // out projection + residual
        const int dt = wave;
        v8f c = wmma_gemm(qs, 128, wo + dt * 16, 128, 128, v8f{}, l16, half);
#pragma unroll
        for (int vv = 0; vv < 8; ++vv) {
            int r = vv + 8 * half, col = dt * 16 + l16;
            ys[r * 128 + col] = c[vv] + bo[col] + xs[r * 128 + col];
        }
    }
    __syncthreads();
    if (t < 13) {
        float mu = 0.f;
        for (int d = 0; d < 128; ++d) mu += ys[t * 128 + d];
        mu *= (1.f / 128.f);
        float var = 0.f;
        for (int d = 0; d < 128; ++d) { float dd = ys[t * 128 + d] - mu; var += dd * dd; }
        float rstd = rsqrtf(var * (1.f / 128.f) + 1e-5f);
        for (int d = 0; d < 128; ++d)
            out[(size_t)bs * cND + t * 128 + d] =
                (ys[t * 128 + d] - mu) * rstd * g[d] + bb[d];
    }
}

// ---------------------------------------------------------------------------
// decoder: adj = tanh(gelu(enr@w1+b1)@w2+b2)
__global__ void k_dec(const float* __restrict__ enr, const float* __restrict__ w1,
                      const float* __restrict__ b1, const float* __restrict__ w2,
                      const float* __restrict__ b2, float* __restrict__ adj) {
    int e = blockIdx.x * blockDim.x + threadIdx.x;
    if (e >= cNE) return;
    const float* er = enr + (size_t)e * 8;
    float y = b2[0];
#pragma unroll
    for (int j = 0; j < 4; ++j) {
        float acc = b1[j];
#pragma unroll
        for (int k = 0; k < 8; ++k) acc += er[k] * w1[k * 4 + j];
        y += gelu_erf(acc) * w2[j];
    }
    adj[e] = tanhf(y);
}

// ---------------------------------------------------------------------------
extern "C" void kernel_launch(void* const* d_in, const int* in_sizes, int n_in,
                              void* d_out, int out_size, void* d_ws, size_t ws_size,
                              hipStream_t stream) {
    (void)in_sizes; (void)n_in; (void)out_size; (void)ws_size;
    const float* ast = (const float*)d_in[0];
    const float* lrn = (const float*)d_in[1];
    const float* att = (const float*)d_in[2];
    const float* enc = (const float*)d_in[3];
    // params in setup_inputs() insertion order
    auto P = [&](int i) { return (const float*)d_in[4 + i]; };
    // enc: 0 w1,1 b1,2 ln1_g,3 ln1_b,4 w2,5 b2,6 ln2_g,7 ln2_b
    // fusion: 8 w1_ctx,9 w1_edge,10 b1,11 w2,12 b2
    // mp[0]: 13..23  mp[1]: 24..34  (w_msg,b_msg,wg_i,wg_j,wg_e,bg,wu_n,wu_m,bu,ln_g,ln_b)
    // temporal: 35..44 (wq,bq,wk,bk,wv,bv,wo,bo,ln_g,ln_b)  spatial: 45..54  dec: 55..58
    float* ws = (float*)d_ws;
    float* enr = ws;                      // cNE*8
    float* ctx = enr + (size_t)cNE * 8;   // cBS*16
    float* ns  = ctx + (size_t)cBS * 16;  // cNSD
    float* qb  = ns + (size_t)cNSD;       // cNSD  ([bn][s][128])
    float* kb  = qb + (size_t)cNSD;
    float* vb  = kb + (size_t)cNSD;
    float* ao  = vb + (size_t)cNSD;
    float* adj_out = (float*)d_out;
    float* ns_out  = adj_out + cNE;

    k_ctx<<<(cBS * 16 + 255) / 256, 256, 0, stream>>>(enc, P(8), ctx);
    k_edge<<<(cNE + 255) / 256, 256, 0, stream>>>(
        ast, lrn, att, ctx, P(0), P(1), P(2), P(3), P(4), P(5), P(6), P(7),
        P(9), P(10), P(11), P(12), enr);
    k_init_ns<<<(cNSD + 255) / 256, 256, 0, stream>>>(enc, ns);
    for (int r = 0; r < 2; ++r) {
        int o = 13 + r * 11;
        k_mp<<<cBS, 256, 0, stream>>>(enr, ns, P(o), P(o + 1), P(o + 2), P(o + 3),
                                      P(o + 4), P(o + 5), P(o + 6), P(o + 7),
                                      P(o + 8), P(o + 9), P(o + 10));
    }
    // temporal MHA
    k_tqkv<<<cBN * 32, 256, 0, stream>>>(ns, P(35), P(36), P(37), P(38), P(39),
                                         P(40), qb, kb, vb);
    k_tattn<<<cBN * cH, 256, 0, stream>>>(qb, kb, vb, ao);
    k_tout<<<cBN * 32, 256, 0, stream>>>(ao, ns, P(41), P(42), P(43), P(44));
    // spatial MHA -> final node states straight into d_out
    k_spatial<<<cBS, 256, 0, stream>>>(ns, P(45), P(46), P(47), P(48), P(49),
                                       P(50), P(51), P(52), P(53), P(54), ns_out);
    // edge decoder -> adjacency in d_out
    k_dec<<<(cNE + 255) / 256, 256, 0, stream>>>(enr, P(55), P(56), P(57), P(58),
                                                 adj_out);
}